// MeanTokenProjectionPool_9431748182742
// MI455X (gfx1250) — compile-verified
//
#include <hip/hip_runtime.h>
#include <hip/hip_bf16.h>

// Problem constants (match reference)
#define B_   32
#define T_   4096
#define D_   512
#define G_   8
#define OUT_ 512

#define TS_     32            // t-slices for the reduction
#define TCHUNK  (T_ / TS_)    // 128 tokens per block

#define SUMS_ELEMS (G_ * B_ * D_)   // 131072 floats, laid out [G][B][D]
#define CNTS_ELEMS (B_ * G_)        // 256 floats, laid out [B][G]

typedef float v2f __attribute__((ext_vector_type(2)));
typedef float v8f __attribute__((ext_vector_type(8)));

// ---------------------------------------------------------------- init
__global__ void mtpp_init_ws(float* __restrict__ ws) {
    int i = blockIdx.x * blockDim.x + threadIdx.x;
    if (i < SUMS_ELEMS + CNTS_ELEMS) ws[i] = 0.0f;
}

// ---------------------------------------------------------------- counts
// counts[b][g] = # of valid (non-padded) tokens of type g in batch row b.
__global__ void mtpp_count(const int* __restrict__ types,
                           const unsigned char* __restrict__ mask,
                           float* __restrict__ counts) {
    const int b = blockIdx.x;
    float c0=0.f,c1=0.f,c2=0.f,c3=0.f,c4=0.f,c5=0.f,c6=0.f,c7=0.f;
    for (int t = threadIdx.x; t < T_; t += blockDim.x) {
        if (mask[b * T_ + t]) continue;   // padded
        const int g = types[t];
        c0 += (g == 0); c1 += (g == 1); c2 += (g == 2); c3 += (g == 3);
        c4 += (g == 4); c5 += (g == 5); c6 += (g == 6); c7 += (g == 7);
    }
    float* cb = counts + b * G_;
    unsafeAtomicAdd(cb + 0, c0); unsafeAtomicAdd(cb + 1, c1);
    unsafeAtomicAdd(cb + 2, c2); unsafeAtomicAdd(cb + 3, c3);
    unsafeAtomicAdd(cb + 4, c4); unsafeAtomicAdd(cb + 5, c5);
    unsafeAtomicAdd(cb + 6, c6); unsafeAtomicAdd(cb + 7, c7);
}

// ---------------------------------------------------------------- reduction
// Segmented sum: sums[g][b][d] += batch[b][t][d] over valid t of type g.
// token type & padding bit are wave-uniform -> scalar branches; padded rows
// are skipped before the load (halves HBM traffic).
__global__ void mtpp_sum(const float* __restrict__ batch,
                         const int* __restrict__ types,
                         const unsigned char* __restrict__ mask,
                         float* __restrict__ sums) {
    const int b  = blockIdx.x;
    const int t0 = blockIdx.y * TCHUNK;
    const int d  = threadIdx.x * 4;             // blockDim.x = D_/4 = 128

    float4 a0 = {0,0,0,0}, a1 = {0,0,0,0}, a2 = {0,0,0,0}, a3 = {0,0,0,0};
    float4 a4 = {0,0,0,0}, a5 = {0,0,0,0}, a6 = {0,0,0,0}, a7 = {0,0,0,0};

    const float* bp = batch + (size_t)b * T_ * D_ + d;
    const unsigned char* mrow = mask + (size_t)b * T_;

    for (int t = t0; t < t0 + TCHUNK; ++t) {
        if (mrow[t]) continue;                  // uniform: skip padded row
        const int g = types[t];                 // uniform: scalar switch below
        const float4 v = *(const float4*)(bp + (size_t)t * D_);
        switch (g) {
            case 0: a0.x+=v.x; a0.y+=v.y; a0.z+=v.z; a0.w+=v.w; break;
            case 1: a1.x+=v.x; a1.y+=v.y; a1.z+=v.z; a1.w+=v.w; break;
            case 2: a2.x+=v.x; a2.y+=v.y; a2.z+=v.z; a2.w+=v.w; break;
            case 3: a3.x+=v.x; a3.y+=v.y; a3.z+=v.z; a3.w+=v.w; break;
            case 4: a4.x+=v.x; a4.y+=v.y; a4.z+=v.z; a4.w+=v.w; break;
            case 5: a5.x+=v.x; a5.y+=v.y; a5.z+=v.z; a5.w+=v.w; break;
            case 6: a6.x+=v.x; a6.y+=v.y; a6.z+=v.z; a6.w+=v.w; break;
            default:a7.x+=v.x; a7.y+=v.y; a7.z+=v.z; a7.w+=v.w; break;
        }
    }

#define MTPP_STORE_G(gi, acc) {                                              \
        float* p = sums + ((size_t)((gi) * B_ + b) * D_) + d;                \
        unsafeAtomicAdd(p + 0, (acc).x); unsafeAtomicAdd(p + 1, (acc).y);    \
        unsafeAtomicAdd(p + 2, (acc).z); unsafeAtomicAdd(p + 3, (acc).w); }
    MTPP_STORE_G(0, a0); MTPP_STORE_G(1, a1); MTPP_STORE_G(2, a2);
    MTPP_STORE_G(3, a3); MTPP_STORE_G(4, a4); MTPP_STORE_G(5, a5);
    MTPP_STORE_G(6, a6); MTPP_STORE_G(7, a7);
#undef MTPP_STORE_G
}

// ---------------------------------------------------------------- WMMA GEMM
// out[b][g][o] = (sums[g][b][:] * scale_b) @ W[g][:][o] + bias[g][o]
// One wave per 16x16 output tile; V_WMMA_F32_16X16X4_F32, K-loop over D.
// A (16x4 f32) layout: lanes 0-15 = M, VGPR0 holds K0 (lo half)/K2 (hi half),
// VGPR1 holds K1/K3.  C/D (16x16 f32): VGPR r -> M=r (lanes 0-15), M=r+8
// (lanes 16-31), N = lane&15.
__global__ void __launch_bounds__(32)
mtpp_gemm(const float* __restrict__ W,
          const float* __restrict__ bias,
          const float* __restrict__ sums,
          const float* __restrict__ counts,
          float* __restrict__ out) {
    const int g     = blockIdx.x / (B_ / 16);
    const int mtile = blockIdx.x % (B_ / 16);
    const int ntile = blockIdx.y;

    const int lane = threadIdx.x;
    const int m    = lane & 15;
    const int hi   = lane >> 4;          // 0: K0/K1 rows, 1: K2/K3 rows
    const int brow = mtile * 16 + m;

    const float cnt   = counts[brow * G_ + g];
    const float scale = cnt > 0.0f ? 1.0f / cnt : 0.0f;   // empty group -> 0

    const float* arow = sums + (size_t)(g * B_ + brow) * D_ + hi * 2;
    const float* wcol = W + (size_t)g * D_ * OUT_ + ntile * 16 + m;

    v8f c = {0.f,0.f,0.f,0.f,0.f,0.f,0.f,0.f};
    for (int k0 = 0; k0 < D_; k0 += 4) {
        v2f a;
        a.x = arow[k0 + 0] * scale;
        a.y = arow[k0 + 1] * scale;
        const float* wk = wcol + (size_t)(k0 + hi * 2) * OUT_;
        v2f bb;
        bb.x = wk[0];
        bb.y = wk[OUT_];
        // 8 args: (neg_a, A, neg_b, B, c_mod, C, reuse_a, reuse_b)
        c = __builtin_amdgcn_wmma_f32_16x16x4_f32(
                false, a, false, bb, (short)0, c, false, false);
    }

    const int ocol = ntile * 16 + m;
    const float bv = bias[g * OUT_ + ocol];
#pragma unroll
    for (int r = 0; r < 8; ++r) {
        const int bo = mtile * 16 + r + hi * 8;
        out[((size_t)bo * G_ + g) * OUT_ + ocol] = c[r] + bv;
    }
}

// ---------------------------------------------------------------- launch
extern "C" void kernel_launch(void* const* d_in, const int* in_sizes, int n_in,
                              void* d_out, int out_size, void* d_ws, size_t ws_size,
                              hipStream_t stream) {
    const float*         batch = (const float*)d_in[0];          // [B,T,D] f32
    const float*         W     = (const float*)d_in[1];          // [G,D,OUT] f32
    const float*         bbias = (const float*)d_in[2];          // [G,OUT] f32
    const int*           types = (const int*)d_in[3];            // [T] i32
    const unsigned char* mask  = (const unsigned char*)d_in[4];  // [B,T] bool
    float*               out   = (float*)d_out;                  // [B,G,OUT] f32

    float* sums   = (float*)d_ws;            // [G][B][D]
    float* counts = sums + SUMS_ELEMS;       // [B][G]   (513 KB total scratch)

    mtpp_init_ws<<<(SUMS_ELEMS + CNTS_ELEMS + 255) / 256, 256, 0, stream>>>(sums);
    mtpp_count<<<B_, 256, 0, stream>>>(types, mask, counts);
    mtpp_sum<<<dim3(B_, TS_), D_ / 4, 0, stream>>>(batch, types, mask, sums);
    mtpp_gemm<<<dim3(G_ * (B_ / 16), OUT_ / 16), 32, 0, stream>>>(W, bbias, sums, counts, out);
}